// SparseSubdivideBlock3d_14465449853509
// MI455X (gfx1250) — compile-verified
//
#include <hip/hip_runtime.h>
#include <hip/hip_bf16.h>
#include <math.h>

typedef __attribute__((ext_vector_type(2))) float v2f;
typedef __attribute__((ext_vector_type(8))) float v8f;

#define NPTS    20000
#define NCHILD  (NPTS * 8)          // 160000
#define CIN0    128
#define COUT    64
#define NGROUP  32
#define GDIM    66                  // RES_OUT + 2
#define GVOX    (GDIM * GDIM * GDIM)
#define EPSGN   1e-5f

__device__ __forceinline__ float silu_(float x) { return x / (1.0f + __expf(-x)); }

// ---------------------------------------------------------------------------
// GroupNorm statistics: one block per group, reduce sum / sumsq over
// nrows * CPG elements (reference normalizes over axes (0,2) jointly).
// Writes stats[2g] = mean, stats[2g+1] = rsqrt(var + eps).
// ---------------------------------------------------------------------------
template <int CTOT, int CPG>
__global__ void gn_stats_kernel(const float* __restrict__ x, int nrows,
                                float* __restrict__ stats) {
    const int g  = blockIdx.x;
    const int c0 = g * CPG;
    const int total = nrows * CPG;
    float s = 0.0f, s2 = 0.0f;
    for (int e = threadIdx.x; e < total; e += 256) {
        int row = e / CPG;
        int c   = e - row * CPG;
        float v = x[(size_t)row * CTOT + c0 + c];
        s += v; s2 += v * v;
    }
    __shared__ float ls[256], lq[256];
    ls[threadIdx.x] = s; lq[threadIdx.x] = s2;
    __syncthreads();
    for (int off = 128; off > 0; off >>= 1) {
        if (threadIdx.x < off) {
            ls[threadIdx.x] += ls[threadIdx.x + off];
            lq[threadIdx.x] += lq[threadIdx.x + off];
        }
        __syncthreads();
    }
    if (threadIdx.x == 0) {
        float inv = 1.0f / (float)total;
        float mu  = ls[0] * inv;
        float var = lq[0] * inv - mu * mu;
        stats[2 * g]     = mu;
        stats[2 * g + 1] = rsqrtf(var + EPSGN);
    }
}

// ---------------------------------------------------------------------------
// Apply GroupNorm affine + SiLU elementwise.
// ---------------------------------------------------------------------------
template <int CTOT, int CPG>
__global__ void gn_silu_kernel(const float* __restrict__ x,
                               const float* __restrict__ stats,
                               const float* __restrict__ gamma,
                               const float* __restrict__ beta,
                               float* __restrict__ out, long n) {
    long i = (long)blockIdx.x * blockDim.x + threadIdx.x;
    if (i >= n) return;
    int c = (int)(i % CTOT);
    int g = c / CPG;
    float y = (x[i] - stats[2 * g]) * stats[2 * g + 1] * gamma[c] + beta[c];
    out[i] = silu_(y);
}

// ---------------------------------------------------------------------------
// Voxel index grid: -1 = empty, else child point id (0..NCHILD-1).
// ---------------------------------------------------------------------------
__global__ void init_grid_kernel(int* __restrict__ grid) {
    int i = blockIdx.x * blockDim.x + threadIdx.x;
    if (i < GVOX) grid[i] = -1;
}

__global__ void build_grid_kernel(const int* __restrict__ coords,
                                  int* __restrict__ grid) {
    int m = blockIdx.x * blockDim.x + threadIdx.x;
    if (m >= NCHILD) return;
    int n = m >> 3, o = m & 7;
    int gx = 2 * coords[4 * n + 1] + ((o >> 2) & 1) + 1;
    int gy = 2 * coords[4 * n + 2] + ((o >> 1) & 1) + 1;
    int gz = 2 * coords[4 * n + 3] + (o & 1) + 1;
    grid[(gx * GDIM + gy) * GDIM + gz] = m;
}

// ---------------------------------------------------------------------------
// Gather-GEMM 3x3x3 submanifold conv via V_WMMA_F32_16X16X4_F32.
// One wave computes a 16-point x 64-channel output tile (4 x v8f accum).
//
// Per-tap weights are staged in LDS in K-pair-swizzled layout:
//     wtap[(kk>>1)*COUT*2 + col*2 + (kk&1)] = W[kk][col]
// so a B fragment (W[kk][col], W[kk+1][col]) is ONE aligned ds_load_b64
// (kk = k + koff is always even) — no VALU repacking between DS and WMMA.
//
// PARENT_FEATS: conv1 reads parent feature row (child>>3).
// ADD_SKIP:     conv2 fuses out += skip[child>>3].
// ---------------------------------------------------------------------------
template <int CINT, bool PARENT_FEATS, bool ADD_SKIP>
__global__ void conv_wmma_kernel(const int* __restrict__ coords,
                                 const int* __restrict__ grid,
                                 const float* __restrict__ feats,
                                 const float* __restrict__ W,
                                 const float* __restrict__ bias,
                                 const float* __restrict__ skip,
                                 float* __restrict__ out) {
    __shared__ float wtap[CINT * COUT];
    const int lane = threadIdx.x & 31;
    const int wave = threadIdx.x >> 5;
    const int tile = blockIdx.x * 4 + wave;          // 10000 tiles, exact fit

    // Per-lane output row (M = lane&15 of the tile) -> base voxel in grid.
    const int mrow = tile * 16 + (lane & 15);
    const int n = mrow >> 3, o = mrow & 7;
    const int gx = 2 * coords[4 * n + 1] + ((o >> 2) & 1) + 1;
    const int gy = 2 * coords[4 * n + 2] + ((o >> 1) & 1) + 1;
    const int gz = 2 * coords[4 * n + 3] + (o & 1) + 1;

    v8f acc[4] = {};
    const int koff  = (lane < 16) ? 0 : 2;           // A-frag K lanes (ISA 7.12.2)
    const int col15 = lane & 15;

    for (int t = 0; t < 27; ++t) {
        __syncthreads();
        const float* wsrc = W + (size_t)t * CINT * COUT;
        // Stage tap weights, swizzling into K-pair-major layout.
        for (int e = threadIdx.x; e < CINT * COUT; e += 128) {
            int row = e / COUT;              // kk (cin index)
            int col = e - row * COUT;        // cout index
            wtap[((row >> 1) * COUT + col) * 2 + (row & 1)] = wsrc[e];
        }
        if (t < 26)                           // hide next tap's fetch latency
            __builtin_prefetch(wsrc + CINT * COUT, 0, 3);
        __syncthreads();

        const int di = t / 9 - 1, dj = (t / 3) % 3 - 1, dk = t % 3 - 1;
        const int nidx = grid[((gx + di) * GDIM + (gy + dj)) * GDIM + (gz + dk)];
        const float* arow = nullptr;
        if (nidx >= 0) {
            int frow = PARENT_FEATS ? (nidx >> 3) : nidx;
            arow = feats + (size_t)frow * CINT;
        }

        for (int k = 0; k < CINT; k += 4) {
            v2f a = {0.0f, 0.0f};
            if (arow) a = *(const v2f*)(arow + k + koff);
            const float* wp = wtap + ((k + koff) >> 1) * (COUT * 2);
#pragma unroll
            for (int nt = 0; nt < 4; ++nt) {
                v2f b = *(const v2f*)(wp + (nt * 16 + col15) * 2);
                acc[nt] = __builtin_amdgcn_wmma_f32_16x16x4_f32(
                    false, a, false, b, (short)0, acc[nt], false, false);
            }
        }
    }

    // Bias + (skip) + store.  C/D layout: vgpr r, lanes<16 -> M=r, lanes>=16 -> M=8+r.
    const int rowlo = tile * 16 + ((lane < 16) ? 0 : 8);
#pragma unroll
    for (int nt = 0; nt < 4; ++nt) {
        const int col = nt * 16 + col15;
        const float bb = bias[col];
#pragma unroll
        for (int r = 0; r < 8; ++r) {
            const int row = rowlo + r;
            float v = acc[nt][r] + bb;
            if (ADD_SKIP) v += skip[(size_t)(row >> 3) * COUT + col];
            out[(size_t)row * COUT + col] = v;
        }
    }
}

// ---------------------------------------------------------------------------
// Skip path: skip[n] = x_feats[n] @ Ws + bs  (N x 128 @ 128 x 64), WMMA f32.
// One wave per 16-row tile; Ws staged once in LDS (same K-pair swizzle).
// ---------------------------------------------------------------------------
__global__ void skip_gemm_kernel(const float* __restrict__ X,
                                 const float* __restrict__ Ws,
                                 const float* __restrict__ bs,
                                 float* __restrict__ out) {
    __shared__ float wl[CIN0 * COUT];
    const int lane = threadIdx.x & 31;
    const int wave = threadIdx.x >> 5;
    for (int e = threadIdx.x; e < CIN0 * COUT; e += 128) {
        int row = e / COUT;
        int col = e - row * COUT;
        wl[((row >> 1) * COUT + col) * 2 + (row & 1)] = Ws[e];
    }
    __syncthreads();

    const int tile = blockIdx.x * 4 + wave;
    if (tile * 16 >= NPTS) return;                   // whole wave exits together

    const int col15 = lane & 15;
    const int koff  = (lane < 16) ? 0 : 2;
    const float* arow = X + (size_t)(tile * 16 + col15) * CIN0;

    v8f acc[4] = {};
    for (int k = 0; k < CIN0; k += 4) {
        v2f a = *(const v2f*)(arow + k + koff);
        const float* wp = wl + ((k + koff) >> 1) * (COUT * 2);
#pragma unroll
        for (int nt = 0; nt < 4; ++nt) {
            v2f b = *(const v2f*)(wp + (nt * 16 + col15) * 2);
            acc[nt] = __builtin_amdgcn_wmma_f32_16x16x4_f32(
                false, a, false, b, (short)0, acc[nt], false, false);
        }
    }

    const int rowlo = tile * 16 + ((lane < 16) ? 0 : 8);
#pragma unroll
    for (int nt = 0; nt < 4; ++nt) {
        const int col = nt * 16 + col15;
        const float bb = bs[col];
#pragma unroll
        for (int r = 0; r < 8; ++r)
            out[(size_t)(rowlo + r) * COUT + col] = acc[nt][r] + bb;
    }
}

// ---------------------------------------------------------------------------
// Host-side launch.
// ---------------------------------------------------------------------------
extern "C" void kernel_launch(void* const* d_in, const int* in_sizes, int n_in,
                              void* d_out, int out_size, void* d_ws, size_t ws_size,
                              hipStream_t stream) {
    const float* x_feats = (const float*)d_in[0];
    const int*   x_coords = (const int*)d_in[1];
    const float* gn1_g = (const float*)d_in[2];
    const float* gn1_b = (const float*)d_in[3];
    const float* W1    = (const float*)d_in[4];
    const float* b1    = (const float*)d_in[5];
    const float* gn2_g = (const float*)d_in[6];
    const float* gn2_b = (const float*)d_in[7];
    const float* W2    = (const float*)d_in[8];
    const float* b2    = (const float*)d_in[9];
    const float* Ws    = (const float*)d_in[10];
    const float* bs    = (const float*)d_in[11];
    float* outp = (float*)d_out;

    // Workspace carve-up (256B aligned), total ~99 MB.
    size_t off = 0;
    auto carve = [&](size_t bytes) -> char* {
        char* p = (char*)d_ws + off;
        off += (bytes + 255) & ~(size_t)255;
        return p;
    };
    float* h1     = (float*)carve((size_t)NPTS * CIN0 * 4);       // silu(gn1(x))
    float* out1   = (float*)carve((size_t)NCHILD * COUT * 4);     // conv1 result
    float* h2     = (float*)carve((size_t)NCHILD * COUT * 4);     // silu(gn2(conv1))
    float* skipb  = (float*)carve((size_t)NPTS * COUT * 4);       // x @ Ws + bs
    float* stats1 = (float*)carve(2 * NGROUP * 4);
    float* stats2 = (float*)carve(2 * NGROUP * 4);
    int*   grid   = (int*)carve((size_t)GVOX * 4);
    (void)ws_size; (void)in_sizes; (void)n_in; (void)out_size;

    // 1) GN1 stats + apply + SiLU
    gn_stats_kernel<CIN0, 4><<<NGROUP, 256, 0, stream>>>(x_feats, NPTS, stats1);
    {
        long n = (long)NPTS * CIN0;
        gn_silu_kernel<CIN0, 4><<<(unsigned)((n + 255) / 256), 256, 0, stream>>>(
            x_feats, stats1, gn1_g, gn1_b, h1, n);
    }

    // 2) Build voxel index grid for the subdivided cloud.
    init_grid_kernel<<<(GVOX + 255) / 256, 256, 0, stream>>>(grid);
    build_grid_kernel<<<(NCHILD + 255) / 256, 256, 0, stream>>>(x_coords, grid);

    // 3) Skip GEMM (per-parent; broadcast to children at conv2).
    skip_gemm_kernel<<<(NPTS / 16 + 3) / 4, 128, 0, stream>>>(x_feats, Ws, bs, skipb);

    // 4) Conv1: gather-GEMM, parent-replicated features.
    conv_wmma_kernel<CIN0, true, false><<<NCHILD / 16 / 4, 128, 0, stream>>>(
        x_coords, grid, h1, W1, b1, nullptr, out1);

    // 5) GN2 stats + apply + SiLU
    gn_stats_kernel<COUT, 2><<<NGROUP, 256, 0, stream>>>(out1, NCHILD, stats2);
    {
        long n = (long)NCHILD * COUT;
        gn_silu_kernel<COUT, 2><<<(unsigned)((n + 255) / 256), 256, 0, stream>>>(
            out1, stats2, gn2_g, gn2_b, h2, n);
    }

    // 6) Conv2 fused with skip add, writes d_out directly.
    conv_wmma_kernel<COUT, false, true><<<NCHILD / 16 / 4, 128, 0, stream>>>(
        x_coords, grid, h2, W2, b2, skipb, outp);
}